// PaperSimpleGC_69475390980641
// MI455X (gfx1250) — compile-verified
//
#include <hip/hip_runtime.h>
#include <hip/hip_bf16.h>
#include <math.h>

#define N_NODES 20000
#define N_EDGES 640000
#define K_CHEB  20
#define BATCH   128
#define NPB     40   // nodes per block in fused kernel
#define NCH     8    // nodes staged in LDS per chunk

typedef __attribute__((ext_vector_type(2))) float v2f;
typedef __attribute__((ext_vector_type(8))) float v8f;

// ---------------- CSR build ----------------

__global__ void k_zero(int* __restrict__ cnt, float* __restrict__ h_part) {
    int i = blockIdx.x * 256 + threadIdx.x;
    if (i < N_NODES) cnt[i] = 0;
    if (i < BATCH * 16) h_part[i] = 0.0f;
}

__global__ void k_count(const int* __restrict__ rows, int* __restrict__ cnt) {
    int e = blockIdx.x * 256 + threadIdx.x;
    if (e < N_EDGES) atomicAdd(&cnt[rows[e]], 1);
}

// single-block exclusive scan: counts (in 'fill') -> row_ptr, and reset fill=row start cursors
__global__ void k_scan(int* __restrict__ fill, int* __restrict__ row_ptr) {
    __shared__ int s[256];
    __shared__ int running;
    int t = threadIdx.x;
    if (t == 0) running = 0;
    __syncthreads();
    for (int base = 0; base < N_NODES; base += 256) {
        int i = base + t;
        int v = (i < N_NODES) ? fill[i] : 0;
        s[t] = v;
        __syncthreads();
        for (int off = 1; off < 256; off <<= 1) {
            int add = (t >= off) ? s[t - off] : 0;
            __syncthreads();
            s[t] += add;
            __syncthreads();
        }
        int excl = running + s[t] - v;
        if (i < N_NODES) { row_ptr[i] = excl; fill[i] = excl; }
        int tot = s[255];
        __syncthreads();
        if (t == 0) running += tot;
        __syncthreads();
    }
    if (t == 0) row_ptr[N_NODES] = running;
}

__global__ void k_scatter(const int* __restrict__ rows, const int* __restrict__ cols,
                          const float* __restrict__ vals, int* __restrict__ fill,
                          int* __restrict__ col_s, float* __restrict__ val_s) {
    int e = blockIdx.x * 256 + threadIdx.x;
    if (e < N_EDGES) {
        int r = rows[e];
        int p = atomicAdd(&fill[r], 1);
        col_s[p] = cols[e];
        val_s[p] = vals[e];
    }
}

// ---------------- Chebyshev recursion ----------------

// T0[n][b] = x[b][n]
__global__ void k_init_T0(const float* __restrict__ x, float* __restrict__ T0) {
    int n = blockIdx.x;
    int b = threadIdx.x;
    T0[n * BATCH + b] = x[(size_t)b * N_NODES + n];
}

// Tout = (scale2 ? 2*L*Tprev - Tprev2 : L*Tprev); one block per row, thread = batch lane
__global__ void k_spmm(const int* __restrict__ rp, const int* __restrict__ ci,
                       const float* __restrict__ cv,
                       const float* __restrict__ Tprev, const float* __restrict__ Tprev2,
                       float* __restrict__ Tout, int scale2) {
    int n = blockIdx.x;
    int b = threadIdx.x;
    int e0 = rp[n], e1 = rp[n + 1];
    float acc = 0.0f;
    for (int e = e0; e < e1; ++e) {
        int   c = ci[e];   // uniform across block -> scalarized
        float v = cv[e];
        acc = fmaf(v, Tprev[c * BATCH + b], acc);
    }
    float r = scale2 ? (2.0f * acc - Tprev2[n * BATCH + b]) : acc;
    Tout[n * BATCH + b] = r;
}

// ---------------- Fused gc (WMMA) + relu + FC (WMMA) ----------------
// gc[n,b,f] = relu( sum_k Xt[k][n][b]*Wgc[k][f] + bgc[f] )   (16x16x4 f32 WMMA, K=20 in 5 chunks)
// h[b,f']  += sum_{n,f} gc[n,b,f] * Wfc[n*10+f][f']          (16x16x4 f32 WMMA, A staged via LDS)
__global__ void k_fused(const float* __restrict__ Xt, const float* __restrict__ Wgc,
                        const float* __restrict__ bgc, const float* __restrict__ Wfc,
                        float* __restrict__ h_part) {
    __shared__ float Rld[NCH * 10][BATCH];   // 40 KB relu(gc) staging

    const int tid  = threadIdx.x;
    const int lane = tid & 31;
    const int wave = tid >> 5;       // 4 waves
    const int hf   = lane >> 4;      // half-wave index
    const int fl   = lane & 15;      // N-column within 16x16 tile
    const int nbase = blockIdx.x * NPB;

    // B operand of gc GEMM: Wgc chunks (constant over n) -- zero-pad f>=10
    v2f Bg[5];
#pragma unroll
    for (int kc = 0; kc < 5; ++kc) {
        int kk = kc * 4 + 2 * hf;
        v2f b;
        b[0] = (fl < 10) ? Wgc[kk * 10 + fl] : 0.0f;
        b[1] = (fl < 10) ? Wgc[(kk + 1) * 10 + fl] : 0.0f;
        Bg[kc] = b;
    }
    const float bias = (fl < 10) ? bgc[fl] : 0.0f;

    // FC accumulators: wave owns b-tiles {2*wave, 2*wave+1}; persist across whole n loop
    v8f C2[2];
    C2[0] = (v8f){0,0,0,0,0,0,0,0};
    C2[1] = (v8f){0,0,0,0,0,0,0,0};

    for (int nb = 0; nb < NPB; nb += NCH) {
        // ---- phase 1: gc tiles -> relu -> LDS ----
        for (int ln = 0; ln < NCH; ++ln) {
            int n = nbase + nb + ln;
#pragma unroll
            for (int t = 0; t < 2; ++t) {
                int b0 = wave * 32 + t * 16;
                v8f D = (v8f){0,0,0,0,0,0,0,0};
#pragma unroll
                for (int kc = 0; kc < 5; ++kc) {
                    int kk = kc * 4 + 2 * hf;
                    v2f A;
                    A[0] = Xt[(size_t)(kk * N_NODES + n) * BATCH + b0 + fl];
                    A[1] = Xt[(size_t)((kk + 1) * N_NODES + n) * BATCH + b0 + fl];
                    D = __builtin_amdgcn_wmma_f32_16x16x4_f32(
                            false, A, false, Bg[kc], (short)0, D, false, false);
                }
                if (fl < 10) {
                    int row = ln * 10 + fl;
                    int bb  = b0 + 8 * hf;          // contiguous 8 b's per lane
                    float4 p0, p1;
                    p0.x = fmaxf(D[0] + bias, 0.0f);
                    p0.y = fmaxf(D[1] + bias, 0.0f);
                    p0.z = fmaxf(D[2] + bias, 0.0f);
                    p0.w = fmaxf(D[3] + bias, 0.0f);
                    p1.x = fmaxf(D[4] + bias, 0.0f);
                    p1.y = fmaxf(D[5] + bias, 0.0f);
                    p1.z = fmaxf(D[6] + bias, 0.0f);
                    p1.w = fmaxf(D[7] + bias, 0.0f);
                    *(float4*)&Rld[row][bb]     = p0;
                    *(float4*)&Rld[row][bb + 4] = p1;
                }
            }
        }
        __syncthreads();

        // ---- phase 2: FC GEMM over the 80 staged rows ----
        int rowg = (nbase + nb) * 10;   // global Wfc row base for this chunk
#pragma unroll 4
        for (int kc2 = 0; kc2 < (NCH * 10) / 4; ++kc2) {
            int k0 = kc2 * 4;
            int kk = k0 + 2 * hf;
            v2f Bf;
            Bf[0] = (fl < 10) ? Wfc[(rowg + kk) * 10 + fl] : 0.0f;
            Bf[1] = (fl < 10) ? Wfc[(rowg + kk + 1) * 10 + fl] : 0.0f;
#pragma unroll
            for (int t = 0; t < 2; ++t) {
                int m0 = wave * 32 + t * 16;
                v2f A;
                A[0] = Rld[kk][m0 + fl];
                A[1] = Rld[kk + 1][m0 + fl];
                C2[t] = __builtin_amdgcn_wmma_f32_16x16x4_f32(
                            false, A, false, Bf, (short)0, C2[t], false, false);
            }
        }
        __syncthreads();
    }

    // ---- flush FC accumulators ----
#pragma unroll
    for (int t = 0; t < 2; ++t) {
        if (fl < 10) {
            int bb = wave * 32 + t * 16 + 8 * hf;
#pragma unroll
            for (int v = 0; v < 8; ++v)
                atomicAdd(&h_part[(bb + v) * 16 + fl], C2[t][v]);
        }
    }
}

// ---------------- bias + relu + softmax ----------------
__global__ void k_softmax(const float* __restrict__ h_part, const float* __restrict__ bfc,
                          float* __restrict__ out) {
    int b = threadIdx.x;   // 128
    float hv[10];
    float mx = -INFINITY;
#pragma unroll
    for (int f = 0; f < 10; ++f) {
        float v = fmaxf(h_part[b * 16 + f] + bfc[f], 0.0f);
        hv[f] = v;
        mx = fmaxf(mx, v);
    }
    float s = 0.0f;
#pragma unroll
    for (int f = 0; f < 10; ++f) {
        hv[f] = __expf(hv[f] - mx);
        s += hv[f];
    }
    float inv = 1.0f / s;
#pragma unroll
    for (int f = 0; f < 10; ++f)
        out[b * 10 + f] = hv[f] * inv;
}

// ---------------- host launcher ----------------

extern "C" void kernel_launch(void* const* d_in, const int* in_sizes, int n_in,
                              void* d_out, int out_size, void* d_ws, size_t ws_size,
                              hipStream_t stream) {
    const float* x    = (const float*)d_in[0];   // [128, 20000, 1]
    const int*   rows = (const int*)  d_in[1];   // [640000]
    const int*   cols = (const int*)  d_in[2];   // [640000]
    const float* vals = (const float*)d_in[3];   // [640000]
    const float* Wgc  = (const float*)d_in[4];   // [20, 10]
    const float* bgc  = (const float*)d_in[5];   // [10]
    const float* Wfc  = (const float*)d_in[6];   // [200000, 10]
    const float* bfc  = (const float*)d_in[7];   // [10]
    float* out = (float*)d_out;                  // [128, 10]

    // workspace layout
    const size_t NB = (size_t)N_NODES * BATCH;           // elems per T_k
    float* Xt      = (float*)d_ws;                       // K_CHEB * NB floats (204.8 MB)
    int*   row_ptr = (int*)(Xt + (size_t)K_CHEB * NB);   // N+1
    int*   fill    = row_ptr + (N_NODES + 1);            // N (counts, then cursors)
    int*   col_s   = fill + N_NODES;                     // E
    float* val_s   = (float*)(col_s + N_EDGES);          // E
    float* h_part  = val_s + N_EDGES;                    // 128*16

    // CSR build
    k_zero   <<<(N_NODES + 255) / 256, 256, 0, stream>>>(fill, h_part);
    k_count  <<<(N_EDGES + 255) / 256, 256, 0, stream>>>(rows, fill);
    k_scan   <<<1, 256, 0, stream>>>(fill, row_ptr);
    k_scatter<<<(N_EDGES + 255) / 256, 256, 0, stream>>>(rows, cols, vals, fill, col_s, val_s);

    // Chebyshev recursion into Xt[k] slots
    k_init_T0<<<N_NODES, BATCH, 0, stream>>>(x, Xt);
    k_spmm   <<<N_NODES, BATCH, 0, stream>>>(row_ptr, col_s, val_s, Xt, Xt, Xt + NB, 0);
    for (int k = 2; k < K_CHEB; ++k)
        k_spmm<<<N_NODES, BATCH, 0, stream>>>(row_ptr, col_s, val_s,
                                              Xt + (size_t)(k - 1) * NB,
                                              Xt + (size_t)(k - 2) * NB,
                                              Xt + (size_t)k * NB, 1);

    // fused WMMA gc + FC
    k_fused<<<N_NODES / NPB, 128, 0, stream>>>(Xt, Wgc, bgc, Wfc, h_part);

    // bias + relu + softmax
    k_softmax<<<1, BATCH, 0, stream>>>(h_part, bfc, out);
}